// ConvLRU_78176994722068
// MI455X (gfx1250) — compile-verified
//
#include <hip/hip_runtime.h>
#include <hip/hip_bf16.h>
#include <math.h>

// ---------------------------------------------------------------------------
// ConvLRU on MI455X (gfx1250): freq-domain LRU block.
// Channel-axis FFT folded into the channel-mix matrices so both channel
// stages become complex GEMMs (M=262144,N=64,K=64) on v_wmma_f32_16x16x4_f32.
// Global->LDS staging uses gfx1250 async-to-LDS ops (ASYNCcnt), with a
// double-buffered X-slab pipeline in the GEMM kernel.
// ---------------------------------------------------------------------------

#define EMB_CH 64
#define HH     32
#define WW     32
#define BSZ    2
#define LSZ    128
#define HWN    1024                 // HH*WW
#define PLANES (BSZ*LSZ*EMB_CH)     // 16384 complex (h,w) planes
#define BLN    (BSZ*LSZ)            // 256
#define NBUF   ((size_t)BSZ*LSZ*EMB_CH*HWN)   // 16,777,216 floats per field
#define LN_EPS 1e-5f
#define TWO_PI 6.2831853071795864f

typedef __attribute__((ext_vector_type(2))) float v2f;
typedef __attribute__((ext_vector_type(8))) float v8f;

__device__ __forceinline__ int rev5(int i) {
    return ((i & 1) << 4) | ((i & 2) << 2) | (i & 4) | ((i & 8) >> 2) | ((i & 16) >> 4);
}

// LDS byte offset of a __shared__ object: flat LDS aperture keeps the
// workgroup-relative offset in addr[31:0].
__device__ __forceinline__ unsigned lds_off(const void* p) {
    return (unsigned)(unsigned long long)p;
}

// gfx1250 async global->LDS copies (tracked by ASYNCcnt).
__device__ __forceinline__ void async_b128(unsigned lds, const float* g) {
    asm volatile("global_load_async_to_lds_b128 %0, %1, off"
                 :: "v"(lds), "v"((unsigned long long)g) : "memory");
}
__device__ __forceinline__ void async_b32(unsigned lds, const float* g) {
    asm volatile("global_load_async_to_lds_b32 %0, %1, off"
                 :: "v"(lds), "v"((unsigned long long)g) : "memory");
}
__device__ __forceinline__ void wait_async0() {
    asm volatile("s_wait_asynccnt 0x0" ::: "memory");
}

// ---------------------------------------------------------------------------
// Kernel A: precompute Wb' = Wb @ F64, Wc' = (1/64) Wc @ conj(F64),
//           lambda = exp(-exp(nu_log) + i*exp(theta_log)), gamma = exp(gamma_log)
// ---------------------------------------------------------------------------
__global__ __launch_bounds__(256) void k_precompute(
    const float* __restrict__ Wb_r, const float* __restrict__ Wb_i,
    const float* __restrict__ Wc_r, const float* __restrict__ Wc_i,
    const float* __restrict__ plog,
    float* __restrict__ Wbp_r, float* __restrict__ Wbp_i,
    float* __restrict__ Wcp_r, float* __restrict__ Wcp_i,
    float* __restrict__ lamr, float* __restrict__ lami, float* __restrict__ gam)
{
    int t = blockIdx.x * 256 + threadIdx.x;      // 0..4095
    int o = t >> 6, n = t & 63;
    float br = 0.f, bi = 0.f, cr = 0.f, ci = 0.f;
    for (int k = 0; k < 64; ++k) {
        float ang = -TWO_PI * (float)((k * n) & 63) * (1.0f / 64.0f);
        float s, c;
        __sincosf(ang, &s, &c);                  // fwd twiddle = c + i s
        float wr = Wb_r[o * 64 + k], wi = Wb_i[o * 64 + k];
        br += wr * c - wi * s;
        bi += wr * s + wi * c;
        float vr = Wc_r[o * 64 + k], vi = Wc_i[o * 64 + k];   // inv twiddle = c - i s
        cr += vr * c + vi * s;
        ci += vi * c - vr * s;
    }
    Wbp_r[t] = br;  Wbp_i[t] = bi;
    Wcp_r[t] = cr * (1.0f / 64.0f);
    Wcp_i[t] = ci * (1.0f / 64.0f);

    if (t < EMB_CH * HH) {                       // t = c*32 + h
        float nu = __expf(plog[t]);
        float th = __expf(plog[2048 + t]);
        float e  = __expf(-nu);
        lamr[t] = e * __cosf(th);
        lami[t] = e * __sinf(th);
        gam[t]  = __expf(plog[4096 + t]);
    }
}

// ---------------------------------------------------------------------------
// Kernel B: forward 2D FFT over (H,W), real input -> complex (split r/i).
// One 32x32 plane per workgroup, radix-2 DIT in LDS, 256 threads.
// The bit-reversal permutation along w happens inside the async-to-LDS
// scatter (per-lane LDS destination addresses).
// ---------------------------------------------------------------------------
__global__ __launch_bounds__(256) void k_fft2d_fwd(
    const float* __restrict__ x, float* __restrict__ outr, float* __restrict__ outi)
{
    __shared__ __align__(16) float ar[1024], ai[1024], br_[1024], bi_[1024];
    const int tid = threadIdx.x;
    const size_t plane = (size_t)blockIdx.x * HWN;
    const unsigned ar_b = lds_off(ar);

    for (int e = tid; e < 1024; e += 256) {          // async load, bit-reverse w
        int h = e >> 5, w = e & 31;
        async_b32(ar_b + (unsigned)(h * 32 + rev5(w)) * 4u, x + plane + e);
        ai[h * 32 + rev5(w)] = 0.f;
    }
    wait_async0();
    __syncthreads();
    for (int s = 1; s <= 5; ++s) {                   // row FFTs (along w)
        int m = 1 << s, half = m >> 1;
        for (int j = tid; j < 512; j += 256) {
            int row = j >> 4, k = j & 15;
            int grp = k / half, pos = k - grp * half;
            int i1 = row * 32 + grp * m + pos, i2 = i1 + half;
            float ang = -TWO_PI * (float)pos / (float)m;
            float tw_s, tw_c; __sincosf(ang, &tw_s, &tw_c);
            float vr = ar[i2] * tw_c - ai[i2] * tw_s;
            float vi = ar[i2] * tw_s + ai[i2] * tw_c;
            float ur = ar[i1], ui = ai[i1];
            ar[i1] = ur + vr; ai[i1] = ui + vi;
            ar[i2] = ur - vr; ai[i2] = ui - vi;
        }
        __syncthreads();
    }
    for (int e = tid; e < 1024; e += 256) {          // bit-reverse h into b
        int h = e >> 5, w = e & 31;
        br_[rev5(h) * 32 + w] = ar[e];
        bi_[rev5(h) * 32 + w] = ai[e];
    }
    __syncthreads();
    for (int s = 1; s <= 5; ++s) {                   // column FFTs (along h)
        int m = 1 << s, half = m >> 1;
        for (int j = tid; j < 512; j += 256) {
            int col = j >> 4, k = j & 15;
            int grp = k / half, pos = k - grp * half;
            int r1 = grp * m + pos, r2 = r1 + half;
            int i1 = r1 * 32 + col, i2 = r2 * 32 + col;
            float ang = -TWO_PI * (float)pos / (float)m;
            float tw_s, tw_c; __sincosf(ang, &tw_s, &tw_c);
            float vr = br_[i2] * tw_c - bi_[i2] * tw_s;
            float vi = br_[i2] * tw_s + bi_[i2] * tw_c;
            float ur = br_[i1], ui = bi_[i1];
            br_[i1] = ur + vr; bi_[i1] = ui + vi;
            br_[i2] = ur - vr; bi_[i2] = ui - vi;
        }
        __syncthreads();
    }
    for (int e = tid; e < 1024; e += 256) {
        outr[plane + e] = br_[e];
        outi[plane + e] = bi_[e];
    }
}

// ---------------------------------------------------------------------------
// Kernel C/E: complex GEMM per (b,l): D(64x1024) = W'(64x64) x X(64x1024),
// via v_wmma_f32_16x16x4_f32 (4 real products; -Wi negated in registers since
// F32 WMMA only supports C-negate). mode=1 fuses (+bias)*gamma epilogue.
// Block = 256 thr = 8 waves; each wave owns one 16(o) x 16(hw) output tile.
// Each block processes 4 column slabs of 32 (W' loaded once, X slabs
// double-buffered with async-to-LDS prefetch).
// ---------------------------------------------------------------------------
__global__ __launch_bounds__(256) void k_cgemm64(
    const float* __restrict__ Wr, const float* __restrict__ Wi,
    const float* __restrict__ Xr, const float* __restrict__ Xi,
    float* __restrict__ Yr, float* __restrict__ Yi,
    const float* __restrict__ biasr, const float* __restrict__ biasi,
    const float* __restrict__ gam, int mode)
{
    __shared__ __align__(16) float sWr[4096], sWi[4096];       // 64x64 complex W'
    __shared__ __align__(16) float sXr[2][2048], sXi[2][2048]; // double-buffered slabs
    const int tid = threadIdx.x;
    const int bl = blockIdx.y;
    const size_t xbase = (size_t)bl * EMB_CH * HWN;
    const int col0 = blockIdx.x * 128;               // this block: 4 slabs of 32

    // async stage W' (once per block): 16 floats/thread/field = 4 x b128
    {
        const unsigned wr_b = lds_off(sWr), wi_b = lds_off(sWi);
#pragma unroll
        for (int q = 0; q < 4; ++q) {
            async_b128(wr_b + (unsigned)(tid * 64 + q * 16), Wr + tid * 16 + q * 4);
            async_b128(wi_b + (unsigned)(tid * 64 + q * 16), Wi + tid * 16 + q * 4);
        }
    }
    // async stage one 64x32 X slab into buffer `buf`
    auto load_slab = [&](int buf, int colbase) {
        const unsigned xr_b = lds_off(&sXr[buf][0]);
        const unsigned xi_b = lds_off(&sXi[buf][0]);
        const int e0 = tid * 8;
#pragma unroll
        for (int q = 0; q < 2; ++q) {
            int e = e0 + q * 4;                      // 4-float chunk, col-aligned
            int c = e >> 5, col = e & 31;
            const float* gr = Xr + xbase + (size_t)c * HWN + colbase + col;
            const float* gi = Xi + xbase + (size_t)c * HWN + colbase + col;
            async_b128(xr_b + (unsigned)e * 4u, gr);
            async_b128(xi_b + (unsigned)e * 4u, gi);
        }
    };
    load_slab(0, col0);
    wait_async0();
    __syncthreads();

    const int wave = tid >> 5, lane = tid & 31;
    const int otile = wave & 3;              // 4 tiles cover o=0..63
    const int ntile = wave >> 2;             // 2 tiles cover 32 cols
    const int lh = lane >> 4, lm = lane & 15;
    const int arow = otile * 16 + lm;        // A 16x4: lane%16 = M row
    const int bcol = ntile * 16 + lm;        // B 4x16: lane%16 = N col
    const size_t ybase = (size_t)bl * EMB_CH * HWN;

    for (int s = 0; s < 4; ++s) {
        const int colbase = col0 + s * 32;
        if (s < 3) load_slab((s + 1) & 1, colbase + 32);   // prefetch next slab
        const float* cXr = &sXr[s & 1][0];
        const float* cXi = &sXi[s & 1][0];

        v8f accR = {}; v8f accI = {};
#pragma unroll
        for (int k0 = 0; k0 < 64; k0 += 4) {
            int ka = k0 + lh * 2;            // upper lane-half holds K+2..K+3
            v2f ar2, ai2, ain2, br2, bi2;
            ar2[0] = sWr[arow * 64 + ka];   ar2[1] = sWr[arow * 64 + ka + 1];
            ai2[0] = sWi[arow * 64 + ka];   ai2[1] = sWi[arow * 64 + ka + 1];
            ain2[0] = -ai2[0];              ain2[1] = -ai2[1];
            br2[0] = cXr[ka * 32 + bcol];   br2[1] = cXr[(ka + 1) * 32 + bcol];
            bi2[0] = cXi[ka * 32 + bcol];   bi2[1] = cXi[(ka + 1) * 32 + bcol];
            // Dr += Wr*Xr - Wi*Xi ; Di += Wr*Xi + Wi*Xr
            accR = __builtin_amdgcn_wmma_f32_16x16x4_f32(false, ar2,  false, br2, (short)0, accR, false, false);
            accR = __builtin_amdgcn_wmma_f32_16x16x4_f32(false, ain2, false, bi2, (short)0, accR, false, false);
            accI = __builtin_amdgcn_wmma_f32_16x16x4_f32(false, ar2,  false, bi2, (short)0, accI, false, false);
            accI = __builtin_amdgcn_wmma_f32_16x16x4_f32(false, ai2,  false, br2, (short)0, accI, false, false);
        }

        // D layout: VGPR v -> M = v + 8*(lane>=16); N = lane&15
        const int col = colbase + ntile * 16 + lm;
#pragma unroll
        for (int v = 0; v < 8; ++v) {
            int o = otile * 16 + v + 8 * lh;
            float yr = accR[v], yi = accI[v];
            if (mode) {
                int h = col >> 5;
                float g = gam[o * 32 + h];
                yr = (yr + biasr[o]) * g;
                yi = (yi + biasi[o]) * g;
            }
            Yr[ybase + (size_t)o * HWN + col] = yr;
            Yi[ybase + (size_t)o * HWN + col] = yi;
        }
        wait_async0();          // prefetched slab resident
        __syncthreads();        // all waves done reading current slab
    }
}

// ---------------------------------------------------------------------------
// Kernel D: sequential scan over L (in-place), 131072 independent complex
// recurrences s = lambda[c,h]*s + y.  Also emits last hidden (complex64 flat).
// ---------------------------------------------------------------------------
__global__ __launch_bounds__(256) void k_scan(
    float* __restrict__ Yr, float* __restrict__ Yi,
    const float* __restrict__ lamr, const float* __restrict__ lami,
    float* __restrict__ last_out)
{
    int idx = blockIdx.x * 256 + threadIdx.x;     // 0..131071 = (b, c, h, w)
    int b = idx >> 16;
    int chw = idx & 65535;
    int c = chw >> 10;
    int h = (chw >> 5) & 31;
    float lr = lamr[c * 32 + h], li = lami[c * 32 + h];
    size_t off = (size_t)b * LSZ * 65536 + chw;
    float sr = 0.f, si = 0.f;
    for (int l = 0; l < LSZ; ++l) {
        float yr = Yr[off], yi = Yi[off];
        float nr = lr * sr - li * si + yr;
        float ni = lr * si + li * sr + yi;
        sr = nr; si = ni;
        Yr[off] = sr; Yi[off] = si;
        off += 65536;
    }
    last_out[2 * idx]     = sr;   // complex64 interleave, shape (B,1,C,H,W)
    last_out[2 * idx + 1] = si;
}

// ---------------------------------------------------------------------------
// Kernel F: inverse 2D FFT over (H,W), complex in -> real out (scaled 1/1024),
// fused + bc_r[o].
// ---------------------------------------------------------------------------
__global__ __launch_bounds__(256) void k_ifft2d_real(
    const float* __restrict__ inr, const float* __restrict__ ini,
    const float* __restrict__ bc_r, float* __restrict__ outre)
{
    __shared__ __align__(16) float ar[1024], ai[1024], br_[1024], bi_[1024];
    const int tid = threadIdx.x;
    const size_t plane = (size_t)blockIdx.x * HWN;
    const int o = blockIdx.x & 63;                 // plane = bl*64 + o
    const unsigned ar_b = lds_off(ar), ai_b = lds_off(ai);

    for (int e = tid; e < 1024; e += 256) {
        int h = e >> 5, w = e & 31;
        unsigned d = (unsigned)(h * 32 + rev5(w)) * 4u;
        async_b32(ar_b + d, inr + plane + e);
        async_b32(ai_b + d, ini + plane + e);
    }
    wait_async0();
    __syncthreads();
    for (int s = 1; s <= 5; ++s) {                 // inverse row FFTs (+ sign)
        int m = 1 << s, half = m >> 1;
        for (int j = tid; j < 512; j += 256) {
            int row = j >> 4, k = j & 15;
            int grp = k / half, pos = k - grp * half;
            int i1 = row * 32 + grp * m + pos, i2 = i1 + half;
            float ang = TWO_PI * (float)pos / (float)m;
            float tw_s, tw_c; __sincosf(ang, &tw_s, &tw_c);
            float vr = ar[i2] * tw_c - ai[i2] * tw_s;
            float vi = ar[i2] * tw_s + ai[i2] * tw_c;
            float ur = ar[i1], ui = ai[i1];
            ar[i1] = ur + vr; ai[i1] = ui + vi;
            ar[i2] = ur - vr; ai[i2] = ui - vi;
        }
        __syncthreads();
    }
    for (int e = tid; e < 1024; e += 256) {
        int h = e >> 5, w = e & 31;
        br_[rev5(h) * 32 + w] = ar[e];
        bi_[rev5(h) * 32 + w] = ai[e];
    }
    __syncthreads();
    for (int s = 1; s <= 5; ++s) {                 // inverse column FFTs
        int m = 1 << s, half = m >> 1;
        for (int j = tid; j < 512; j += 256) {
            int col = j >> 4, k = j & 15;
            int grp = k / half, pos = k - grp * half;
            int r1 = grp * m + pos, r2 = r1 + half;
            int i1 = r1 * 32 + col, i2 = r2 * 32 + col;
            float ang = TWO_PI * (float)pos / (float)m;
            float tw_s, tw_c; __sincosf(ang, &tw_s, &tw_c);
            float vr = br_[i2] * tw_c - bi_[i2] * tw_s;
            float vi = br_[i2] * tw_s + bi_[i2] * tw_c;
            float ur = br_[i1], ui = bi_[i1];
            br_[i1] = ur + vr; bi_[i1] = ui + vi;
            br_[i2] = ur - vr; bi_[i2] = ui - vi;
        }
        __syncthreads();
    }
    const float bias = bc_r[o];
    for (int e = tid; e < 1024; e += 256) {
        outre[plane + e] = br_[e] * (1.0f / 1024.0f) + bias;
    }
}

// ---------------------------------------------------------------------------
// Kernel G: per-(b,l) mean / rstd over C*H*W = 65536 values.
// ---------------------------------------------------------------------------
__global__ __launch_bounds__(256) void k_stats(
    const float* __restrict__ Hre, float* __restrict__ stats)
{
    __shared__ float s1[256], s2[256];
    const int bl = blockIdx.x, tid = threadIdx.x;
    const size_t base = (size_t)bl * 65536;
    float a = 0.f, q = 0.f;
    for (int i = tid; i < 65536; i += 256) {
        float v = Hre[base + i];
        a += v; q += v * v;
    }
    s1[tid] = a; s2[tid] = q;
    __syncthreads();
    for (int s = 128; s > 0; s >>= 1) {
        if (tid < s) { s1[tid] += s1[tid + s]; s2[tid] += s2[tid + s]; }
        __syncthreads();
    }
    if (tid == 0) {
        float mean = s1[0] * (1.0f / 65536.0f);
        float var = s2[0] * (1.0f / 65536.0f) - mean * mean;
        stats[2 * bl]     = mean;
        stats[2 * bl + 1] = rsqrtf(var + LN_EPS);
    }
}

// ---------------------------------------------------------------------------
// Kernel H: LayerNorm + residual -> d_out.
// ---------------------------------------------------------------------------
__global__ __launch_bounds__(256) void k_lnres(
    const float* __restrict__ Hre, const float* __restrict__ stats,
    const float* __restrict__ ln_w, const float* __restrict__ ln_b,
    const float* __restrict__ x, float* __restrict__ out)
{
    size_t idx = (size_t)blockIdx.x * 256 + threadIdx.x;   // 0..16777215
    int bl = (int)(idx >> 16);
    int chw = (int)(idx & 65535);
    float mean = stats[2 * bl], rstd = stats[2 * bl + 1];
    out[idx] = (Hre[idx] - mean) * rstd * ln_w[chw] + ln_b[chw] + x[idx];
}

// ---------------------------------------------------------------------------
extern "C" void kernel_launch(void* const* d_in, const int* in_sizes, int n_in,
                              void* d_out, int out_size, void* d_ws, size_t ws_size,
                              hipStream_t stream)
{
    const float* x    = (const float*)d_in[0];
    const float* plog = (const float*)d_in[1];
    const float* Wb_r = (const float*)d_in[2];
    const float* Wb_i = (const float*)d_in[3];
    const float* bb_r = (const float*)d_in[4];
    const float* bb_i = (const float*)d_in[5];
    const float* Wc_r = (const float*)d_in[6];
    const float* Wc_i = (const float*)d_in[7];
    const float* bc_r = (const float*)d_in[8];
    const float* bc_i = (const float*)d_in[9];
    (void)bc_i; (void)in_sizes; (void)n_in; (void)ws_size; (void)out_size;
    const float* ln_w = (const float*)d_in[10];
    const float* ln_b = (const float*)d_in[11];

    float* out  = (float*)d_out;
    float* last = out + NBUF;                  // (B,1,C,H,W) complex64 tail

    float* ws  = (float*)d_ws;
    float* b0r = ws;
    float* b0i = ws + NBUF;
    float* b1r = ws + 2 * NBUF;
    float* b1i = ws + 3 * NBUF;
    float* cst = ws + 4 * NBUF;
    float* Wbp_r = cst;          float* Wbp_i = cst + 4096;
    float* Wcp_r = cst + 8192;   float* Wcp_i = cst + 12288;
    float* lamr  = cst + 16384;  float* lami  = cst + 18432;
    float* gam   = cst + 20480;  float* stats = cst + 22528;

    k_precompute<<<16, 256, 0, stream>>>(Wb_r, Wb_i, Wc_r, Wc_i, plog,
                                         Wbp_r, Wbp_i, Wcp_r, Wcp_i,
                                         lamr, lami, gam);
    k_fft2d_fwd<<<PLANES, 256, 0, stream>>>(x, b0r, b0i);
    k_cgemm64<<<dim3(8, BLN), 256, 0, stream>>>(Wbp_r, Wbp_i, b0r, b0i,
                                                b1r, b1i, bb_r, bb_i, gam, 1);
    k_scan<<<512, 256, 0, stream>>>(b1r, b1i, lamr, lami, last);
    k_cgemm64<<<dim3(8, BLN), 256, 0, stream>>>(Wcp_r, Wcp_i, b1r, b1i,
                                                b0r, b0i, nullptr, nullptr, nullptr, 0);
    k_ifft2d_real<<<PLANES, 256, 0, stream>>>(b0r, b0i, bc_r, b1r);
    k_stats<<<BLN, 256, 0, stream>>>(b1r, stats);
    k_lnres<<<65536, 256, 0, stream>>>(b1r, stats, ln_w, ln_b, x, out);
}